// AttentionAugmentedConv_66640712565179
// MI455X (gfx1250) — compile-verified
//
#include <hip/hip_runtime.h>
#include <hip/hip_bf16.h>
#include <math.h>

// ---------------------------------------------------------------------------
// AttentionAugmentedConv for MI455X (gfx1250, wave32, WMMA).
// All GEMMs use V_WMMA_F32_16X16X4_F32 (full fp32, matches fp32 reference).
// Register-blocked: multiple 16x16 tiles per wave to amortize fragment loads.
// ---------------------------------------------------------------------------

typedef __attribute__((ext_vector_type(2))) float v2f;
typedef __attribute__((ext_vector_type(8))) float v8f;

#define NH   8
#define DK   64
#define DV   64
#define DKH  8
#define DVH  8
#define HH   32
#define WW   32
#define CIN  256
#define COUT 256
#define CCONV 192   // FILTERS - DV
#define NPIX 1024   // H*W
#define B_   8

__device__ __forceinline__ v8f wmma_f32(v2f a, v2f b, v8f c) {
  // 8 args: (neg_a, A, neg_b, B, c_mod, C, reuse_a, reuse_b)
  return __builtin_amdgcn_wmma_f32_16x16x4_f32(false, a, false, b, (short)0, c,
                                               false, false);
}

// Fragment index helpers (wave32, per CDNA5 ISA 7.12.2):
//  A (16x4 f32): lane holds A[lane&15][k0], A[lane&15][k0+1], k0 = (lane>>4)*2
//  B (4x16 f32): lane holds B[k0][lane&15], B[k0+1][lane&15]
//  C/D (16x16):  vgpr r of lane -> D[(lane>>4)*8 + r][lane&15]

// ---------------------------------------------------------------------------
// Kernel 1: kqv = x @ qkv_w + qkv_b ; scatter to head-major q (scaled), kT, v
// 4 M-tiles x 3 N-tiles per wave.  grid (8192/64, 192/48), block 32
// ---------------------------------------------------------------------------
__global__ __launch_bounds__(32) void qkv_kernel(
    const float* __restrict__ x, const float* __restrict__ qkv_w,
    const float* __restrict__ qkv_b, float* __restrict__ q_ws,
    float* __restrict__ kT_ws, float* __restrict__ v_ws) {
  const int lane = threadIdx.x & 31;
  const int m0 = blockIdx.x * 64;            // pixel tile base (4 tiles)
  const int n0 = blockIdx.y * 48;            // out-channel base (3 tiles)
  const int ar = lane & 15, ak = (lane >> 4) << 1;
  const int bc = lane & 15, bk = (lane >> 4) << 1;

  const float* __restrict__ arow[4];
#pragma unroll
  for (int mt = 0; mt < 4; ++mt)
    arow[mt] = x + (size_t)(m0 + mt * 16 + ar) * CIN;

  v8f acc[4][3] = {};
  for (int k0 = 0; k0 < CIN; k0 += 4) {
    v2f a[4], bf[3];
#pragma unroll
    for (int mt = 0; mt < 4; ++mt)
      a[mt] = {arow[mt][k0 + ak], arow[mt][k0 + ak + 1]};
#pragma unroll
    for (int nt = 0; nt < 3; ++nt) {
      const int nc = n0 + nt * 16 + bc;
      bf[nt] = {qkv_w[(size_t)(k0 + bk) * (2 * DK + DV) + nc],
                qkv_w[(size_t)(k0 + bk + 1) * (2 * DK + DV) + nc]};
    }
#pragma unroll
    for (int mt = 0; mt < 4; ++mt)
#pragma unroll
      for (int nt = 0; nt < 3; ++nt)
        acc[mt][nt] = wmma_f32(a[mt], bf[nt], acc[mt][nt]);
  }

  const int rbase = (lane >> 4) << 3;
#pragma unroll
  for (int nt = 0; nt < 3; ++nt) {
    const int c = n0 + nt * 16 + bc;
    const float bias = qkv_b[c];
#pragma unroll
    for (int mt = 0; mt < 4; ++mt) {
      for (int r = 0; r < 8; ++r) {
        const int p = m0 + mt * 16 + rbase + r;
        const int bidx = p >> 10, hw = p & (NPIX - 1);
        const float val = acc[mt][nt][r] + bias;
        if (c < DK) {                 // k -> transposed [B][NH][DKH][NPIX]
          const int n = c >> 3, d = c & 7;
          kT_ws[(((size_t)(bidx * NH + n)) * DKH + d) * NPIX + hw] = val;
        } else if (c < 2 * DK) {      // q (scaled) -> [B][NH][NPIX][DKH]
          const int cc = c - DK, n = cc >> 3, d = cc & 7;
          q_ws[(((size_t)(bidx * NH + n)) * NPIX + hw) * DKH + d] =
              val * 0.35355339059327373f;  // DKH^-0.5
        } else {                      // v -> [B][NH][NPIX][DVH]
          const int cc = c - 2 * DK, n = cc >> 3, d = cc & 7;
          v_ws[(((size_t)(bidx * NH + n)) * NPIX + hw) * DVH + d] = val;
        }
      }
    }
  }
}

// ---------------------------------------------------------------------------
// Kernel 2: 3x3 SAME conv as implicit GEMM.
// One image row (32 px = 2 M-tiles, wave-uniform h) x 3 N-tiles per wave.
// grid (8192/32, 192/48), block 32
// ---------------------------------------------------------------------------
__global__ __launch_bounds__(32) void conv_kernel(
    const float* __restrict__ x, const float* __restrict__ conv_w,
    const float* __restrict__ conv_b, float* __restrict__ out) {
  const int lane = threadIdx.x & 31;
  const int row0 = blockIdx.x;                 // 0..255 : (b,h)
  const int b = row0 >> 5;
  const int h = row0 & 31;
  const int n0 = blockIdx.y * 48;
  const int ar = lane & 15, ak = (lane >> 4) << 1;
  const int bc = lane & 15, bk = (lane >> 4) << 1;

  v8f acc[2][3] = {};
  for (int dy = -1; dy <= 1; ++dy) {
    const int hh = h + dy;
    if (hh < 0 || hh >= HH) continue;          // wave-uniform skip
    for (int dx = -1; dx <= 1; ++dx) {
      const float* __restrict__ ap[2];
      bool valid[2];
#pragma unroll
      for (int mt = 0; mt < 2; ++mt) {
        const int ww = mt * 16 + ar + dx;      // per-lane halo
        valid[mt] = (ww >= 0) && (ww < WW);
        ap[mt] = x + (size_t)((b * HH + hh) * WW + (valid[mt] ? ww : 0)) * CIN;
      }
      const float* __restrict__ wp =
          conv_w + (size_t)((dy + 1) * 3 + (dx + 1)) * CIN * CCONV;
      for (int k0 = 0; k0 < CIN; k0 += 4) {
        v2f a[2], bf[3];
#pragma unroll
        for (int mt = 0; mt < 2; ++mt) {
          a[mt].x = valid[mt] ? ap[mt][k0 + ak] : 0.0f;
          a[mt].y = valid[mt] ? ap[mt][k0 + ak + 1] : 0.0f;
        }
#pragma unroll
        for (int nt = 0; nt < 3; ++nt) {
          const int nc = n0 + nt * 16 + bc;
          bf[nt] = {wp[(size_t)(k0 + bk) * CCONV + nc],
                    wp[(size_t)(k0 + bk + 1) * CCONV + nc]};
        }
#pragma unroll
        for (int mt = 0; mt < 2; ++mt)
#pragma unroll
          for (int nt = 0; nt < 3; ++nt)
            acc[mt][nt] = wmma_f32(a[mt], bf[nt], acc[mt][nt]);
      }
    }
  }

  const int rbase = (lane >> 4) << 3;
#pragma unroll
  for (int nt = 0; nt < 3; ++nt) {
    const int c = n0 + nt * 16 + bc;
    const float bias = conv_b[c];
#pragma unroll
    for (int mt = 0; mt < 2; ++mt)
      for (int r = 0; r < 8; ++r) {
        const int wq = mt * 16 + rbase + r;
        out[(size_t)(b * NPIX + h * WW + wq) * COUT + c] = acc[mt][nt][r] + bias;
      }
  }
}

// ---------------------------------------------------------------------------
// Kernel 3: flash attention with decomposed 2D relative logits.
// grid (NPIX/16, NH, B), block 32 (one wave per 16-query tile per head)
// logits[i=(x,y)][j=(X,Y)] = q.k + q.rel_w[31+Y-y] + q.rel_h[31+X-x]
// ---------------------------------------------------------------------------
__global__ __launch_bounds__(32) void attn_kernel(
    const float* __restrict__ q_ws, const float* __restrict__ kT_ws,
    const float* __restrict__ v_ws, const float* __restrict__ key_rel_w,
    const float* __restrict__ key_rel_h, float* __restrict__ attn_pre) {
  __shared__ float qs[16][DKH];     // query tile
  __shared__ float rw[16][WW];      // width rel logits per query row
  __shared__ float rh[16][HH];      // height rel logits per query row
  __shared__ float pbuf[16][16];    // P tile staging (D-layout -> A-layout)

  const int lane = threadIdx.x & 31;
  const int n = blockIdx.y, b = blockIdx.z;
  const int bn = b * NH + n;
  const int i0 = blockIdx.x * 16;
  const int qx = i0 >> 5;                       // query row (uniform in tile)
  const int qy0 = i0 & 31;                      // 0 or 16

  for (int t = lane; t < 16 * DKH; t += 32) {
    const int rr = t >> 3, dd = t & 7;
    qs[rr][dd] = q_ws[(((size_t)bn) * NPIX + (i0 + rr)) * DKH + dd];
  }
  __syncthreads();

  {  // each lane fills 16 entries of rw and rh
    const int rr = lane & 15;
    const int hseg = (lane >> 4) << 4;
    const int yq = qy0 + rr;
    for (int u = 0; u < 16; ++u) {
      const int P = hseg + u;  // key Y (for rw) / key X (for rh), 0..31
      const float* __restrict__ pw = key_rel_w + (size_t)(31 + P - yq) * DKH;
      const float* __restrict__ ph = key_rel_h + (size_t)(31 + P - qx) * DKH;
      float sw = 0.f, sh = 0.f;
      for (int d = 0; d < DKH; ++d) {
        sw += qs[rr][d] * pw[d];
        sh += qs[rr][d] * ph[d];
      }
      rw[rr][P] = sw;
      rh[rr][P] = sh;
    }
  }
  __syncthreads();

  const int ar = lane & 15, ak = (lane >> 4) << 1;
  const int bc = lane & 15, bk = (lane >> 4) << 1;
  const int rbase = (lane >> 4) << 3;

  float mrow[8], lrow[8];
  v8f oacc = {};
  for (int r = 0; r < 8; ++r) { mrow[r] = -1e30f; lrow[r] = 0.f; }

  for (int jt = 0; jt < NPIX / 16; ++jt) {
    const int j0 = jt * 16;
    const int X = j0 >> 5;
    const int Y0 = j0 & 31;

    // S = q @ k^T   (16x16, K=DKH=8 -> 2 WMMAs)
    v8f s = {};
#pragma unroll
    for (int kk = 0; kk < DKH; kk += 4) {
      v2f a = {qs[ar][kk + ak], qs[ar][kk + ak + 1]};
      const float* __restrict__ kp =
          kT_ws + (((size_t)bn) * DKH + (kk + bk)) * NPIX + j0 + bc;
      v2f bb = {kp[0], kp[NPIX]};
      s = wmma_f32(a, bb, s);
    }

    // add relative logits, online softmax (rows split by half-wave)
    float pv[8];
#pragma unroll
    for (int r = 0; r < 8; ++r) {
      const int row = rbase + r;
      pv[r] = s[r] + rw[row][Y0 + bc] + rh[row][X];
    }
#pragma unroll
    for (int r = 0; r < 8; ++r) {
      float t = pv[r];
      for (int msk = 1; msk < 16; msk <<= 1)
        t = fmaxf(t, __shfl_xor(t, msk, 32));     // 16-lane (half) reduce
      const float mnew = fmaxf(mrow[r], t);
      const float scale = __expf(mrow[r] - mnew);
      const float p = __expf(pv[r] - mnew);
      float ssum = p;
      for (int msk = 1; msk < 16; msk <<= 1) ssum += __shfl_xor(ssum, msk, 32);
      lrow[r] = lrow[r] * scale + ssum;
      mrow[r] = mnew;
      oacc[r] = oacc[r] * scale;
      pbuf[rbase + r][bc] = p;
    }
    __syncthreads();

    // O += P @ V   (K=16 keys -> 4 WMMAs; V cols 8..15 padded zero)
#pragma unroll
    for (int kk = 0; kk < 16; kk += 4) {
      v2f a = {pbuf[ar][kk + ak], pbuf[ar][kk + ak + 1]};
      v2f bb;
      if (bc < DVH) {
        const float* __restrict__ vp =
            v_ws + (((size_t)bn) * NPIX + j0 + kk + bk) * DVH + bc;
        bb.x = vp[0];
        bb.y = vp[DVH];
      } else {
        bb.x = 0.f;
        bb.y = 0.f;
      }
      oacc = wmma_f32(a, bb, oacc);
    }
    __syncthreads();
  }

  if (bc < DVH) {
    for (int r = 0; r < 8; ++r) {
      const int i = i0 + rbase + r;
      attn_pre[(((size_t)b) * NPIX + i) * DV + n * DVH + bc] =
          oacc[r] / lrow[r];
    }
  }
}

// ---------------------------------------------------------------------------
// Kernel 4: out[:, 192:256] = attn_pre @ attn_w + attn_b
// 4 M-tiles x 4 N-tiles (full width) per wave.  grid (8192/64), block 32
// ---------------------------------------------------------------------------
__global__ __launch_bounds__(32) void proj_kernel(
    const float* __restrict__ attn_pre, const float* __restrict__ attn_w,
    const float* __restrict__ attn_b, float* __restrict__ out) {
  const int lane = threadIdx.x & 31;
  const int m0 = blockIdx.x * 64;
  const int ar = lane & 15, ak = (lane >> 4) << 1;
  const int bc = lane & 15, bk = (lane >> 4) << 1;

  const float* __restrict__ ap[4];
#pragma unroll
  for (int mt = 0; mt < 4; ++mt)
    ap[mt] = attn_pre + (size_t)(m0 + mt * 16 + ar) * DV;

  v8f acc[4][4] = {};
  for (int k0 = 0; k0 < DV; k0 += 4) {
    v2f a[4], bf[4];
#pragma unroll
    for (int mt = 0; mt < 4; ++mt)
      a[mt] = {ap[mt][k0 + ak], ap[mt][k0 + ak + 1]};
#pragma unroll
    for (int nt = 0; nt < 4; ++nt) {
      const int nc = nt * 16 + bc;
      bf[nt] = {attn_w[(size_t)(k0 + bk) * DV + nc],
                attn_w[(size_t)(k0 + bk + 1) * DV + nc]};
    }
#pragma unroll
    for (int mt = 0; mt < 4; ++mt)
#pragma unroll
      for (int nt = 0; nt < 4; ++nt)
        acc[mt][nt] = wmma_f32(a[mt], bf[nt], acc[mt][nt]);
  }

  const int rbase = (lane >> 4) << 3;
#pragma unroll
  for (int nt = 0; nt < 4; ++nt) {
    const int c = nt * 16 + bc;
    const float bias = attn_b[c];
#pragma unroll
    for (int mt = 0; mt < 4; ++mt)
      for (int r = 0; r < 8; ++r) {
        const int p = m0 + mt * 16 + rbase + r;
        out[(size_t)p * COUT + CCONV + c] = acc[mt][nt][r] + bias;
      }
  }
}

// ---------------------------------------------------------------------------
extern "C" void kernel_launch(void* const* d_in, const int* in_sizes, int n_in,
                              void* d_out, int out_size, void* d_ws,
                              size_t ws_size, hipStream_t stream) {
  const float* x         = (const float*)d_in[0];
  const float* conv_w    = (const float*)d_in[1];
  const float* conv_b    = (const float*)d_in[2];
  const float* qkv_w     = (const float*)d_in[3];
  const float* qkv_b     = (const float*)d_in[4];
  const float* attn_w    = (const float*)d_in[5];
  const float* attn_b    = (const float*)d_in[6];
  const float* key_rel_w = (const float*)d_in[7];
  const float* key_rel_h = (const float*)d_in[8];
  float* out = (float*)d_out;

  // workspace: q | kT | v | attn_pre  (each 512K floats = 2MB; total 8MB)
  const size_t SEG = (size_t)B_ * NH * NPIX * DKH;  // 524288
  float* q_ws      = (float*)d_ws;
  float* kT_ws     = q_ws + SEG;
  float* v_ws      = kT_ws + SEG;
  float* attn_pre  = v_ws + SEG;

  qkv_kernel<<<dim3((B_ * NPIX) / 64, (2 * DK + DV) / 48), 32, 0, stream>>>(
      x, qkv_w, qkv_b, q_ws, kT_ws, v_ws);
  conv_kernel<<<dim3(B_ * HH, CCONV / 48), 32, 0, stream>>>(x, conv_w, conv_b,
                                                            out);
  attn_kernel<<<dim3(NPIX / 16, NH, B_), 32, 0, stream>>>(
      q_ws, kT_ws, v_ws, key_rel_w, key_rel_h, attn_pre);
  proj_kernel<<<dim3((B_ * NPIX) / 64), 32, 0, stream>>>(attn_pre, attn_w,
                                                         attn_b, out);
}